// GCN_77292231458882
// MI455X (gfx1250) — compile-verified
//
#include <hip/hip_runtime.h>
#include <math.h>

typedef float v2f __attribute__((ext_vector_type(2)));
typedef float v8f __attribute__((ext_vector_type(8)));

// ---------- WMMA helpers (V_WMMA_F32_16X16X4_F32, full f32) ----------
static __device__ __forceinline__ v8f wmma4(v2f a, v2f b, v8f c) {
  // 8 args: (neg_a, A, neg_b, B, c_mod, C, reuse_a, reuse_b)
  return __builtin_amdgcn_wmma_f32_16x16x4_f32(false, a, false, b, (short)0, c,
                                               false, false);
}

// A fragment for 16x4 f32: lanes 0-15 -> M=lane, K={k0,k0+1}; lanes 16-31 -> K={k0+2,k0+3}
static __device__ __forceinline__ v2f ldsA_frag(const float* A, int ld, int k0, int lane) {
  int row = lane & 15;
  int kk  = k0 + ((lane >> 4) << 1);
  return *(const v2f*)(A + row * ld + kk);   // ld even, kk even -> 8B aligned
}

// B fragment from pre-packed weights: Wp[((k/2)*npad + n)*2 + (k&1)] = W[k][n]
// -> one 8B load per fragment, lanes 0-15 read 128 contiguous bytes
static __device__ __forceinline__ v2f wBp_frag(const float* __restrict__ Wp, int npad,
                                               int n0, int k0, int lane) {
  int n  = n0 + (lane & 15);
  int kk = k0 + ((lane >> 4) << 1);          // even
  return *(const v2f*)(Wp + ((size_t)(kk >> 1) * npad + n) * 2);
}

// 16xK (LDS) x K x16 (packed global W) -> 16x16 accumulator, chained f32 WMMAs
static __device__ __forceinline__ v8f tile_gemm(const float* A, int ldA, int K,
                                                const float* __restrict__ Wp, int npad,
                                                int n0, int lane) {
  v8f acc = {};
  for (int k0 = 0; k0 < K; k0 += 4)
    acc = wmma4(ldsA_frag(A, ldA, k0, lane), wBp_frag(Wp, npad, n0, k0, lane), acc);
  return acc;
}

// D layout: VGPR v -> row v (lanes 0-15) / v+8 (lanes 16-31), col = n0 + (lane&15)
static __device__ __forceinline__ void store_d_lds(float* D, int ld, int n0, v8f acc,
                                                   const float* __restrict__ bias,
                                                   bool relu, int lane) {
  int col   = n0 + (lane & 15);
  int rbase = (lane >> 4) << 3;
  float bv  = bias[col];
#pragma unroll
  for (int v = 0; v < 8; ++v) {
    float val = acc[v] + bv;
    if (relu) val = fmaxf(val, 0.0f);
    D[(rbase + v) * ld + col] = val;
  }
}

// ---------------------------- kernels ----------------------------
__global__ void zero_kernel(uint4* __restrict__ p, int n4) {
  int i = blockIdx.x * blockDim.x + threadIdx.x;
  if (i < n4) p[i] = make_uint4(0u, 0u, 0u, 0u);
}

// Interleave (k,k+1) pairs: out[((k>>1)*npad + n)*2 + (k&1)] = (n<nout)?W[k][n]:0
__global__ void pack_kernel(const float* __restrict__ W, float* __restrict__ out,
                            int K, int nout, int npad) {
  int t = blockIdx.x * blockDim.x + threadIdx.x;
  if (t >= K * npad) return;
  int k = t / npad, n = t % npad;
  float v = (n < nout) ? W[k * nout + n] : 0.0f;
  out[((size_t)(k >> 1) * npad + n) * 2 + (k & 1)] = v;
}

__global__ void deg_kernel(const int* __restrict__ src, const int* __restrict__ dst,
                           float* deg_out, float* deg_in, int E) {
  int e = blockIdx.x * blockDim.x + threadIdx.x;
  if (e >= E) return;
  unsafeAtomicAdd(&deg_out[src[e]], 1.0f);
  unsafeAtomicAdd(&deg_in[dst[e]], 1.0f);
}

__global__ void norms_kernel(const float* __restrict__ deg_out,
                             const float* __restrict__ deg_in,
                             const int* __restrict__ gids,
                             float* norm_out, float* norm_in, float* counts, int N) {
  int i = blockIdx.x * blockDim.x + threadIdx.x;
  if (i >= N) return;
  float d0 = deg_out[i];
  float d1 = deg_in[i];
  norm_out[i] = (d0 > 0.0f) ? (1.0f / sqrtf(d0)) : 0.0f;
  norm_in[i]  = (d1 > 0.0f) ? (1.0f / sqrtf(d1)) : 0.0f;
  unsafeAtomicAdd(&counts[gids[i]], 1.0f);
}

// agg[dst] += (x * norm_out)[src]; one thread handles 8 features of one edge
template <int D>
__global__ void scatter_kernel(const int* __restrict__ src, const int* __restrict__ dst,
                               const float* __restrict__ x,
                               const float* __restrict__ norm_out,
                               float* agg, int E) {
  constexpr int P = D / 8;
  int t = blockIdx.x * blockDim.x + threadIdx.x;
  int e = t / P;
  if (e >= E) return;
  int part = t % P;
  int s = src[e], d = dst[e];
  float w = norm_out[s];
  const float4* xs = (const float4*)(x + (size_t)s * D + part * 8);
  float4 a = xs[0], b = xs[1];
  float* ag = agg + (size_t)d * D + part * 8;
  unsafeAtomicAdd(ag + 0, a.x * w); unsafeAtomicAdd(ag + 1, a.y * w);
  unsafeAtomicAdd(ag + 2, a.z * w); unsafeAtomicAdd(ag + 3, a.w * w);
  unsafeAtomicAdd(ag + 4, b.x * w); unsafeAtomicAdd(ag + 5, b.y * w);
  unsafeAtomicAdd(ag + 6, b.z * w); unsafeAtomicAdd(ag + 7, b.w * w);
}

// h1 = relu((agg1 * norm_in) @ W1 + b1), 8 -> 16
__global__ void __launch_bounds__(32) gemm1_kernel(const float* __restrict__ agg,
                                                   const float* __restrict__ norm_in,
                                                   const float* __restrict__ W1p,
                                                   const float* __restrict__ bias,
                                                   float* __restrict__ out, int N) {
  const int LDA = 12;
  __shared__ float A[16 * LDA];
  int lane = threadIdx.x;
  int row = lane & 15, half = lane >> 4;
  int node0 = blockIdx.x * 16;
  int node = node0 + row;
  bool full = (node0 + 16) <= N;             // uniform across the wave
  float ni = (node < N) ? norm_in[node] : 0.0f;
  int ldn = min(node, N - 1);
  float4 v = *(const float4*)(agg + (size_t)ldn * 8 + half * 4);
  *(float4*)(A + row * LDA + half * 4) = make_float4(v.x * ni, v.y * ni, v.z * ni, v.w * ni);
  __syncthreads();
  v8f acc = tile_gemm(A, LDA, 8, W1p, 16, 0, lane);
  int col = lane & 15, rbase = half << 3;
  float bv = bias[col];
  if (full) {
#pragma unroll
    for (int vv = 0; vv < 8; ++vv)
      out[(size_t)(node0 + rbase + vv) * 16 + col] = fmaxf(acc[vv] + bv, 0.0f);
  } else {
#pragma unroll
    for (int vv = 0; vv < 8; ++vv) {
      int nd = node0 + rbase + vv;
      if (nd < N) out[(size_t)nd * 16 + col] = fmaxf(acc[vv] + bv, 0.0f);
    }
  }
}

// h2 = relu((agg2 * norm_in) @ W2 + b2), 16 -> 32
__global__ void __launch_bounds__(32) gemm2_kernel(const float* __restrict__ agg,
                                                   const float* __restrict__ norm_in,
                                                   const float* __restrict__ W2p,
                                                   const float* __restrict__ bias,
                                                   float* __restrict__ out, int N) {
  const int LDA = 20;
  __shared__ float A[16 * LDA];
  int lane = threadIdx.x;
  int row = lane & 15, half = lane >> 4;
  int node0 = blockIdx.x * 16;
  int node = node0 + row;
  bool full = (node0 + 16) <= N;
  float ni = (node < N) ? norm_in[node] : 0.0f;
  int ldn = min(node, N - 1);
  const float4* sp = (const float4*)(agg + (size_t)ldn * 16 + half * 8);
  float4 a = sp[0], b = sp[1];
  *(float4*)(A + row * LDA + half * 8)     = make_float4(a.x * ni, a.y * ni, a.z * ni, a.w * ni);
  *(float4*)(A + row * LDA + half * 8 + 4) = make_float4(b.x * ni, b.y * ni, b.z * ni, b.w * ni);
  __syncthreads();
#pragma unroll
  for (int nt = 0; nt < 2; ++nt) {
    v8f acc = tile_gemm(A, LDA, 16, W2p, 32, nt * 16, lane);
    int col = nt * 16 + (lane & 15), rbase = half << 3;
    float bv = bias[col];
    if (full) {
#pragma unroll
      for (int vv = 0; vv < 8; ++vv)
        out[(size_t)(node0 + rbase + vv) * 32 + col] = fmaxf(acc[vv] + bv, 0.0f);
    } else {
#pragma unroll
      for (int vv = 0; vv < 8; ++vv) {
        int nd = node0 + rbase + vv;
        if (nd < N) out[(size_t)nd * 32 + col] = fmaxf(acc[vv] + bv, 0.0f);
      }
    }
  }
}

// Fused: (agg3*norm_in)@W3+b3 -> maxpool(2) -> fc(64,128)relu -> fc(128,64)relu
//        -> fc(64,32)relu -> fc(32,10) -> atomic per-graph sums.  One wave / 16 nodes.
__global__ void __launch_bounds__(32)
fused_kernel(const float* __restrict__ agg3, const float* __restrict__ norm_in,
             const int* __restrict__ gids,
             const float* __restrict__ W3p, const float* __restrict__ b3,
             const float* __restrict__ fW1p, const float* __restrict__ fb1,
             const float* __restrict__ fW2p, const float* __restrict__ fb2,
             const float* __restrict__ fW3p, const float* __restrict__ fb3,
             const float* __restrict__ fW4p, const float* __restrict__ fb4,
             float* sums, int N) {
  const int LD32 = 36, LD64 = 68, LD128 = 132;   // padded, div-by-4, bank-friendly
  __shared__ float A[16 * 132];
  __shared__ float B[16 * 132];
  int lane = threadIdx.x;
  int row = lane & 15, half = lane >> 4;
  int node0 = blockIdx.x * 16;
  int node = node0 + row;
  bool full = (node0 + 16) <= N;
  float ni = (node < N) ? norm_in[node] : 0.0f;
  int ldn = min(node, N - 1);
  const float4* sp = (const float4*)(agg3 + (size_t)ldn * 32 + half * 16);
#pragma unroll
  for (int i = 0; i < 4; ++i) {
    float4 t = sp[i];
    *(float4*)(A + row * LD32 + half * 16 + i * 4) =
        make_float4(t.x * ni, t.y * ni, t.z * ni, t.w * ni);
  }
  __syncthreads();
  // gconv3 linear: 32 -> 128 (+b3, no relu) into B
  for (int nt = 0; nt < 8; ++nt) {
    v8f acc = tile_gemm(A, LD32, 32, W3p, 128, nt * 16, lane);
    store_d_lds(B, LD128, nt * 16, acc, b3, false, lane);
  }
  __syncthreads();
  // MaxPool1d(2): B 16x128 -> A 16x64
  for (int i = 0; i < 32; ++i) {
    int c = half * 32 + i;
    v2f t = *(const v2f*)(B + row * LD128 + 2 * c);
    A[row * LD64 + c] = fmaxf(t.x, t.y);
  }
  __syncthreads();
  // fc1: 64 -> 128, relu
  for (int nt = 0; nt < 8; ++nt) {
    v8f acc = tile_gemm(A, LD64, 64, fW1p, 128, nt * 16, lane);
    store_d_lds(B, LD128, nt * 16, acc, fb1, true, lane);
  }
  __syncthreads();
  // fc2: 128 -> 64, relu
  for (int nt = 0; nt < 4; ++nt) {
    v8f acc = tile_gemm(B, LD128, 128, fW2p, 64, nt * 16, lane);
    store_d_lds(A, LD64, nt * 16, acc, fb2, true, lane);
  }
  __syncthreads();
  // fc3: 64 -> 32, relu
  for (int nt = 0; nt < 2; ++nt) {
    v8f acc = tile_gemm(A, LD64, 64, fW3p, 32, nt * 16, lane);
    store_d_lds(B, LD32, nt * 16, acc, fb3, true, lane);
  }
  __syncthreads();
  // fc4: 32 -> 10 (fW4p zero-padded to 16 cols -> uniform WMMA loop)
  {
    v8f acc = tile_gemm(B, LD32, 32, fW4p, 16, 0, lane);
    int n = lane & 15;
    if (n < 10) {
      float bv = fb4[n];
      int rbase = half << 3;
      if (full) {
#pragma unroll
        for (int vv = 0; vv < 8; ++vv) {
          int g = gids[node0 + rbase + vv];
          unsafeAtomicAdd(&sums[g * 10 + n], acc[vv] + bv);
        }
      } else {
#pragma unroll
        for (int vv = 0; vv < 8; ++vv) {
          int nd = node0 + rbase + vv;
          if (nd < N) {
            int g = gids[nd];
            unsafeAtomicAdd(&sums[g * 10 + n], acc[vv] + bv);
          }
        }
      }
    }
  }
}

__global__ void finalize_kernel(const float* __restrict__ sums,
                                const float* __restrict__ counts,
                                float* __restrict__ out, int total) {
  int i = blockIdx.x * blockDim.x + threadIdx.x;
  if (i >= total) return;
  int g = i / 10;
  out[i] = sums[i] / fmaxf(counts[g], 1.0f);
}

// ---------------------------- launcher ----------------------------
extern "C" void kernel_launch(void* const* d_in, const int* in_sizes, int n_in,
                              void* d_out, int out_size, void* d_ws, size_t ws_size,
                              hipStream_t stream) {
  (void)n_in; (void)ws_size;
  const int*   src    = (const int*)d_in[0];
  const int*   dst    = (const int*)d_in[1];
  const int*   gids   = (const int*)d_in[2];
  const float* n_feat = (const float*)d_in[3];
  const float* W1  = (const float*)d_in[4];
  const float* b1  = (const float*)d_in[5];
  const float* W2  = (const float*)d_in[6];
  const float* b2  = (const float*)d_in[7];
  const float* W3  = (const float*)d_in[8];
  const float* b3  = (const float*)d_in[9];
  const float* fW1 = (const float*)d_in[10];
  const float* fb1 = (const float*)d_in[11];
  const float* fW2 = (const float*)d_in[12];
  const float* fb2 = (const float*)d_in[13];
  const float* fW3w= (const float*)d_in[14];
  const float* fb3 = (const float*)d_in[15];
  const float* fW4 = (const float*)d_in[16];
  const float* fb4 = (const float*)d_in[17];

  const int E = in_sizes[0];
  const int N = in_sizes[2];
  const int G = 256;

  float* ws = (float*)d_ws;
  size_t off = 0;
  float* deg_out = ws + off; off += N;
  float* deg_in  = ws + off; off += N;
  float* agg1    = ws + off; off += (size_t)N * 8;
  float* agg2    = ws + off; off += (size_t)N * 16;
  float* agg3    = ws + off; off += (size_t)N * 32;
  float* sums    = ws + off; off += (size_t)G * 10;
  float* counts  = ws + off; off += G;
  size_t zero_words = off;                 // everything above must start at 0
  float* norm_out = ws + off; off += N;
  float* norm_in  = ws + off; off += N;
  float* h1       = ws + off; off += (size_t)N * 16;
  float* h2       = ws + off; off += (size_t)N * 32;
  // packed weights (pair-interleaved; fW4 zero-padded to 16 cols)
  float* W1p  = ws + off; off += 8 * 16;
  float* W2p  = ws + off; off += 16 * 32;
  float* W3p  = ws + off; off += 32 * 128;
  float* fW1p = ws + off; off += 64 * 128;
  float* fW2p = ws + off; off += 128 * 64;
  float* fW3p = ws + off; off += 64 * 32;
  float* fW4p = ws + off; off += 32 * 16;

  const int tiles = (N + 15) / 16;

  {
    int n4 = (int)((zero_words + 3) / 4);  // may spill into norm_out; overwritten later
    zero_kernel<<<(n4 + 255) / 256, 256, 0, stream>>>((uint4*)ws, n4);
  }
  // pack weights (tiny, L2-resident afterwards)
  pack_kernel<<<(8 * 16 + 255) / 256, 256, 0, stream>>>(W1, W1p, 8, 16, 16);
  pack_kernel<<<(16 * 32 + 255) / 256, 256, 0, stream>>>(W2, W2p, 16, 32, 32);
  pack_kernel<<<(32 * 128 + 255) / 256, 256, 0, stream>>>(W3, W3p, 32, 128, 128);
  pack_kernel<<<(64 * 128 + 255) / 256, 256, 0, stream>>>(fW1, fW1p, 64, 128, 128);
  pack_kernel<<<(128 * 64 + 255) / 256, 256, 0, stream>>>(fW2, fW2p, 128, 64, 64);
  pack_kernel<<<(64 * 32 + 255) / 256, 256, 0, stream>>>(fW3w, fW3p, 64, 32, 32);
  pack_kernel<<<(32 * 16 + 255) / 256, 256, 0, stream>>>(fW4, fW4p, 32, 10, 16);

  deg_kernel<<<(E + 255) / 256, 256, 0, stream>>>(src, dst, deg_out, deg_in, E);
  norms_kernel<<<(N + 255) / 256, 256, 0, stream>>>(deg_out, deg_in, gids,
                                                    norm_out, norm_in, counts, N);
  scatter_kernel<8><<<(E + 255) / 256, 256, 0, stream>>>(src, dst, n_feat, norm_out, agg1, E);
  gemm1_kernel<<<tiles, 32, 0, stream>>>(agg1, norm_in, W1p, b1, h1, N);
  scatter_kernel<16><<<(E * 2 + 255) / 256, 256, 0, stream>>>(src, dst, h1, norm_out, agg2, E);
  gemm2_kernel<<<tiles, 32, 0, stream>>>(agg2, norm_in, W2p, b2, h2, N);
  scatter_kernel<32><<<(E * 4 + 255) / 256, 256, 0, stream>>>(src, dst, h2, norm_out, agg3, E);
  fused_kernel<<<tiles, 32, 0, stream>>>(agg3, norm_in, gids, W3p, b3, fW1p, fb1,
                                         fW2p, fb2, fW3p, fb3, fW4p, fb4, sums, N);
  finalize_kernel<<<(out_size + 255) / 256, 256, 0, stream>>>(sums, counts,
                                                              (float*)d_out, out_size);
}